// GroupedLinear_83425444758208
// MI455X (gfx1250) — compile-verified
//
#include <hip/hip_runtime.h>

typedef __attribute__((ext_vector_type(2))) float v2f;
typedef __attribute__((ext_vector_type(8))) float v8f;

#define Bsz 8192
#define Csz 4096
#define Ksz 5

#define WG_CH 128          // channels per workgroup (8 tiles of 16)
#define WG_ROWS 16         // batch rows per workgroup
#define LDS_COLS 132       // WG_CH + 4 halo (2 each side)
#define LDS_STRIDE 133     // odd stride -> conflict-free column reads
#define BROWS 20           // K-window rows per tile's banded B block

__global__ __launch_bounds__(256)
void banded_wmma_f32_kernel(const float* __restrict__ x,
                            const float* __restrict__ W,
                            const float* __restrict__ bias,
                            float* __restrict__ out)
{
    __shared__ float xs[WG_ROWS * LDS_STRIDE];   // x window  (2128 floats)
    __shared__ float bs[8 * BROWS * 16];         // banded B blocks (2560 floats)

    const int C0 = blockIdx.x * WG_CH;
    const int B0 = blockIdx.y * WG_ROWS;
    const int t  = threadIdx.x;

    // ---- Async-stage x window [B0:B0+16, C0-2:C0+130) directly into LDS. ----
    // Out-of-range columns are clamped (garbage data) and zeroed in a fix-up
    // pass after s_wait_asynccnt, before the barrier.
    for (int e = t; e < WG_ROWS * LDS_COLS; e += 256) {
        int row = e / LDS_COLS;
        int col = e - row * LDS_COLS;
        int cg  = C0 - 2 + col;
        int cgc = min(max(cg, 0), Csz - 1);
        unsigned goff = (unsigned)(((B0 + row) * Csz + cgc) * 4);   // byte offset
        unsigned lds  = (unsigned)(unsigned long long)&xs[row * LDS_STRIDE + col];
        asm volatile("global_load_async_to_lds_b32 %0, %1, %2 offset:0"
                     :
                     : "v"(lds), "v"(goff), "s"(x)
                     : "memory");
    }

    // ---- Build banded B blocks while the async copies are in flight. ----
    // bs[tile][rw][col] = W[c0+col, rw-col]; window rw in [0,20), tap in [0,5).
    for (int e = t; e < 8 * BROWS * 16; e += 256) {
        int tile = e / (BROWS * 16);
        int rem  = e - tile * (BROWS * 16);
        int rw   = rem >> 4;
        int col  = rem & 15;
        int kk   = rw - col;
        int kkc  = min(max(kk, 0), Ksz - 1);
        float wv = W[(C0 + tile * 16 + col) * Ksz + kkc];
        bs[e] = (kk == kkc) ? wv : 0.0f;
    }

    // Wait for this wave's async LDS writes, then zero the clamped halo slots.
    asm volatile("s_wait_asynccnt 0x0" ::: "memory");
    for (int e = t; e < WG_ROWS * LDS_COLS; e += 256) {
        int row = e / LDS_COLS;
        int col = e - row * LDS_COLS;
        int cg  = C0 - 2 + col;
        if (cg < 0 || cg >= Csz)
            xs[row * LDS_STRIDE + col] = 0.0f;
    }
    __syncthreads();

    // ---- One 16x16 output tile per wave (8 waves -> 128 channels). ----
    const int wave = t >> 5;
    const int lane = t & 31;
    const int col  = lane & 15;   // N (and batch row M for the A fragment)
    const int half = lane >> 4;   // 0: K rows {0,1}; 1: K rows {2,3}
    const int c0   = C0 + wave * 16;

    // Accumulator = bias broadcast (every D element of this lane is c0+col).
    const float bv = bias[c0 + col];
    v8f c;
#pragma unroll
    for (int i = 0; i < 8; ++i) c[i] = bv;

    // Fragment base pointers; all loop offsets are compile-time immediates.
    const float* xrow = &xs[col * LDS_STRIDE + wave * 16 + 2 * half];
    const float* bcol = &bs[wave * (BROWS * 16) + (2 * half) * 16 + col];

    // 5 K-chunks of 4 covering window [c0-2, c0+18): D = A(16x4)*B(4x16) + C
#pragma unroll
    for (int k = 0; k < 5; ++k) {
        v2f a;
        a.x = xrow[4 * k];
        a.y = xrow[4 * k + 1];
        v2f bb;
        bb.x = bcol[(4 * k) * 16];
        bb.y = bcol[(4 * k + 1) * 16];
        c = __builtin_amdgcn_wmma_f32_16x16x4_f32(
                /*neg_a=*/false, a, /*neg_b=*/false, bb,
                /*c_mod=*/(short)0, c, /*reuse_a=*/false, /*reuse_b=*/false);
    }

    // ---- Store D (NT): component v -> row M = v + 8*half, col N = col. ----
#pragma unroll
    for (int v = 0; v < 8; ++v) {
        const int Mrow = v + 8 * half;
        __builtin_nontemporal_store(c[v],
            &out[(long)(B0 + Mrow) * Csz + c0 + col]);
    }
}

extern "C" void kernel_launch(void* const* d_in, const int* in_sizes, int n_in,
                              void* d_out, int out_size, void* d_ws, size_t ws_size,
                              hipStream_t stream) {
    const float* x = (const float*)d_in[0];
    const float* W = (const float*)d_in[1];
    const float* b = (const float*)d_in[2];
    float* out = (float*)d_out;

    dim3 grid(Csz / WG_CH, Bsz / WG_ROWS);  // 32 x 512 workgroups
    banded_wmma_f32_kernel<<<grid, dim3(256), 0, stream>>>(x, W, b, out);
}